// GCNModel_1657857376515
// MI455X (gfx1250) — compile-verified
//
#include <hip/hip_runtime.h>
#include <math.h>

// ---------------------------------------------------------------------------
// GCN basis-decomposition model for MI455X (gfx1250, wave32).
//
// Roofline reasoning:
//  * GEMM part: only B=2 GEMMs of [100000,128]x[128,128] = 6.6 GFLOP total
//    (we fold the S=4 basis combination into the edge kernel since
//    sup_s = c_{s0}*H0 + c_{s1}*H1). Trivial compute; done with
//    V_WMMA_F32_16X16X4_F32 to keep fp32 semantics of the reference.
//  * Edge aggregation: 12.8M edges x (1KB gather + 512B scatter) dominates.
//    H table is [2,100000,128] f32 = 102.4 MB -> FITS in the 192 MB L2
//    (supports table [4,N,128] = 205 MB would not). Random gathers and
//    scatter atomics are therefore served by L2, not HBM.
// ---------------------------------------------------------------------------

typedef __attribute__((ext_vector_type(2))) float v2f;
typedef __attribute__((ext_vector_type(4))) float v4f;
typedef __attribute__((ext_vector_type(8))) float v8f;

#define N_NODES 100000
#define F_DIM   128
#define O_DIM   128
#define S_SUP   4
#define B_BAS   2
#define E_EDGES 3200000

// ---------------------------------------------------------------------------
// Kernel 1: H[b] = features @ W[b]   (b = 0..1), fp32 WMMA 16x16x4.
// One wave computes one 16x16 output tile; K=128 in steps of 4 (32 WMMAs).
// ---------------------------------------------------------------------------
__global__ __launch_bounds__(256)
void gemm_h_kernel(const float* __restrict__ features,
                   const float* __restrict__ W,
                   float* __restrict__ H) {
  const int lane       = threadIdx.x & 31;
  const int waveInBlk  = threadIdx.x >> 5;
  const int w          = blockIdx.x * 8 + waveInBlk;   // 100000 waves total

  // tiles: b (2) x rowTile (6250) x colTile (8); colTile fastest so the
  // 8 waves of a block share the same A rows.
  const int b       = w / (6250 * 8);
  const int rem     = w % (6250 * 8);
  const int rowTile = rem >> 3;
  const int colTile = rem & 7;
  const int m0 = rowTile * 16;
  const int n0 = colTile * 16;

  const int hl = lane & 15;   // 0..15
  const int hi = lane >> 4;   // lane half selects K sub-pair

  const float* A  = features + (long)m0 * F_DIM;
  const float* Bm = W + (long)b * F_DIM * O_DIM + n0;

  v8f acc = {};
  #pragma unroll 4
  for (int k = 0; k < F_DIM; k += 4) {
    // A frag (16x4 f32): lanes 0-15 -> K=k,k+1 ; lanes 16-31 -> K=k+2,k+3
    const float* ap = A + (long)hl * F_DIM + k + hi * 2;
    v2f afrag;
    afrag.x = ap[0];
    afrag.y = ap[1];
    // B frag (4x16 f32): VGPR v, half hi -> K = k + 2*hi + v, N = n0 + hl
    const float* bp = Bm + (long)(k + 2 * hi) * O_DIM + hl;
    v2f bfrag;
    bfrag.x = bp[0];
    bfrag.y = bp[O_DIM];
    acc = __builtin_amdgcn_wmma_f32_16x16x4_f32(
        /*neg_a=*/false, afrag, /*neg_b=*/false, bfrag,
        /*c_mod=*/(short)0, acc, /*reuse_a=*/false, /*reuse_b=*/false);
  }

  // C/D layout: VGPR r -> M = r + 8*hi, N = hl
  float* Hout = H + (long)b * N_NODES * O_DIM + (long)m0 * O_DIM + n0;
  #pragma unroll
  for (int r = 0; r < 8; ++r) {
    Hout[(long)(r + 8 * hi) * O_DIM + hl] = acc[r];
  }
}

// ---------------------------------------------------------------------------
// Kernel 2: zero the output accumulator (vec4 stores).
// ---------------------------------------------------------------------------
__global__ __launch_bounds__(256)
void zero_kernel(float* __restrict__ out, long n4) {
  long i = (long)blockIdx.x * blockDim.x + threadIdx.x;
  if (i < n4) {
    v4f z = {};
    *((v4f*)out + i) = z;
  }
}

// ---------------------------------------------------------------------------
// Kernel 3: edge aggregation. One wave per edge (grid-stride).
//   msg = (c_{s0}*H0[col] + c_{s1}*H1[col]) * val ; out[row] += msg
// 32 lanes x float4 covers the 128 outputs. Gathers/atomics hit L2.
// ---------------------------------------------------------------------------
__global__ __launch_bounds__(256)
void edge_agg_kernel(const float* __restrict__ H,
                     const float* __restrict__ Wc,
                     const int*   __restrict__ rows,
                     const int*   __restrict__ cols,
                     const float* __restrict__ vals,
                     float* __restrict__ out) {
  const int  lane   = threadIdx.x & 31;
  const long wave   = ((long)blockIdx.x * blockDim.x + threadIdx.x) >> 5;
  const long nWaves = ((long)gridDim.x * blockDim.x) >> 5;
  const long total  = (long)S_SUP * E_EDGES;

  const float* H0 = H;
  const float* H1 = H + (long)N_NODES * O_DIM;

  for (long idx = wave; idx < total; idx += nWaves) {
    const int   s   = (int)(idx / E_EDGES);
    const int   row = rows[idx];
    const int   col = cols[idx];
    const float val = vals[idx];
    const float c0  = Wc[s * B_BAS + 0];
    const float c1  = Wc[s * B_BAS + 1];

    const v4f h0 = *(const v4f*)(H0 + (long)col * O_DIM + lane * 4);
    const v4f h1 = *(const v4f*)(H1 + (long)col * O_DIM + lane * 4);
    const v4f msg = (h0 * c0 + h1 * c1) * val;

    float* op = out + (long)row * O_DIM + lane * 4;
    atomicAdd(op + 0, msg.x);
    atomicAdd(op + 1, msg.y);
    atomicAdd(op + 2, msg.z);
    atomicAdd(op + 3, msg.w);
  }
}

// ---------------------------------------------------------------------------
// Kernel 4: in-place tanh on the accumulator.
// ---------------------------------------------------------------------------
__global__ __launch_bounds__(256)
void tanh_kernel(float* __restrict__ out, long n) {
  long i = (long)blockIdx.x * blockDim.x + threadIdx.x;
  if (i < n) out[i] = tanhf(out[i]);
}

// ---------------------------------------------------------------------------
// Host launcher.
// d_in order: features [N*F], W [B*F*O], W_comp [S*B],
//             edge_rows [S*E] i32, edge_cols [S*E] i32, edge_vals [S*E] f32
// d_out: [N*O] f32.  d_ws: H table, needs 2*N*O*4 = 102.4 MB.
// ---------------------------------------------------------------------------
extern "C" void kernel_launch(void* const* d_in, const int* in_sizes, int n_in,
                              void* d_out, int out_size, void* d_ws, size_t ws_size,
                              hipStream_t stream) {
  const float* features = (const float*)d_in[0];
  const float* W        = (const float*)d_in[1];
  const float* Wc       = (const float*)d_in[2];
  const int*   rows     = (const int*)d_in[3];
  const int*   cols     = (const int*)d_in[4];
  const float* vals     = (const float*)d_in[5];
  float* out = (float*)d_out;
  float* H   = (float*)d_ws;   // [B, N, O] f32

  (void)in_sizes; (void)n_in; (void)out_size; (void)ws_size;

  // 1) H[b] = features @ W[b] : 100000 wave-tiles / 8 waves per block
  gemm_h_kernel<<<12500, 256, 0, stream>>>(features, W, H);

  // 2) zero the accumulator: N*O/4 vec4 stores
  const long n4 = (long)N_NODES * O_DIM / 4;  // 3.2M
  zero_kernel<<<(int)((n4 + 255) / 256), 256, 0, stream>>>(out, n4);

  // 3) edge aggregation over S*E = 12.8M edges, grid-stride waves
  edge_agg_kernel<<<8192, 256, 0, stream>>>(H, Wc, rows, cols, vals, out);

  // 4) tanh in place
  const long n = (long)N_NODES * O_DIM;       // 12.8M
  tanh_kernel<<<(int)((n + 255) / 256), 256, 0, stream>>>(out, n);
}